// QuantumAttentionLayer_65481071410055
// MI455X (gfx1250) — compile-verified
//
#include <hip/hip_runtime.h>
#include <hip/hip_bf16.h>
#include <math.h>

typedef __attribute__((ext_vector_type(2))) float v2f;
typedef __attribute__((ext_vector_type(4))) float v4f;
typedef __attribute__((ext_vector_type(8))) float v8f;
typedef __attribute__((ext_vector_type(4))) int   v4i;

// pointer-to-(AS1 v4i) and pointer-to-(AS3 v4i), per the async builtin's signature
typedef __attribute__((address_space(1))) v4i* gv4i_p;
typedef __attribute__((address_space(3))) v4i* lv4i_p;

#define B_ 4
#define T_ 4096
#define E_ 4096
#define H_ 32
#define M_ (B_ * T_) // 16384 rows

#define KT     32            // K-tile staged per async round
#define NT     (E_ / KT)     // 128 tiles
#define ROWPAD 36            // 32 + 4 dword pad: rows stay 16B-aligned, ds reads conflict-free
#define WBUF   (16 * ROWPAD) // floats per wave per buffer

#if __has_builtin(__builtin_amdgcn_global_load_async_to_lds_b128) && \
    __has_builtin(__builtin_amdgcn_s_wait_asynccnt)
#define USE_ASYNC_LDS 1
#else
#define USE_ASYNC_LDS 0
#endif

#if USE_ASYNC_LDS
#define GLDS_B128(g, l)                                  \
    __builtin_amdgcn_global_load_async_to_lds_b128(      \
        (gv4i_p)(g), (lv4i_p)(l), 0, 0)
#endif

// ---------------------------------------------------------------------------
// Kernel 0: swizzle W [E,H] into per-lane fragment order so the GEMM inner
// loop fetches a lane's full B data (b0.x,b0.y,b1.x,b1.y) with ONE b128 load.
// Wr[kg*128 + l*4 + {0,1,2,3}] = W[(kg*4 + (l>>4)*2 + {0,1})*H + (l&15) + {0,16}]
// ---------------------------------------------------------------------------
__global__ __launch_bounds__(256) void qattn_swizzle_W(
    const float* __restrict__ Wm, float* __restrict__ Wr)
{
    const int tid = blockIdx.x * 256 + threadIdx.x; // 0 .. (E_/4)*32 - 1
    const int kg  = tid >> 5;                       // k-group (4 K values)
    const int l   = tid & 31;                       // lane
    const int kk  = kg * 4 + (l >> 4) * 2;
    const int n   = l & 15;
    v4f v;
    v.x = Wm[(size_t)kk * H_ + n];
    v.y = Wm[(size_t)(kk + 1) * H_ + n];
    v.z = Wm[(size_t)kk * H_ + n + 16];
    v.w = Wm[(size_t)(kk + 1) * H_ + n + 16];
    *(v4f*)(Wr + (size_t)tid * 4) = v;
}

// ---------------------------------------------------------------------------
// Kernel 1: scores[m,h] = masked( cos( x[m,:] . W[:,h] + params[h] ) )
// One wave = 16 rows x 32 cols x full K via V_WMMA_F32_16X16X4_F32.
// A frag (16x4 f32): lanes 0-15 -> M=lane, VGPR0/1 = K {0,1}; lanes 16-31 -> K {2,3}.
// B frag (4x16 f32): lanes 0-15 -> N=lane, VGPR0/1 = K {0,1}; lanes 16-31 -> K {2,3}.
// C/D (16x16 f32): VGPR j -> M = j + 8*(lane>=16), N = lane&15.
// Async variant: per-wave TRIPLE-buffered LDS staging of the wave's 16 x-rows
// via GLOBAL_LOAD_ASYNC_TO_LDS_B128 (2 tiles in flight), synchronized only with
// s_wait_asynccnt (ASYNCcnt is per-wave; no workgroup barriers needed).
// ---------------------------------------------------------------------------
#if USE_ASYNC_LDS

__global__ __launch_bounds__(256) void qattn_gemm_cos(
    const float* __restrict__ x, const float* __restrict__ Wm,
    const float* __restrict__ Wr, const float* __restrict__ params,
    const int* __restrict__ mask, float* __restrict__ scores)
{
    __shared__ float lds[8 * 3 * WBUF]; // 54 KB

    const int lane = threadIdx.x & 31;
    const int wave = threadIdx.x >> 5;
    const int half = lane >> 4;   // K pair {0,1} vs {2,3}
    const int n    = lane & 15;   // A: row-in-tile; B/D: column-in-half
    const int row0 = blockIdx.x * 128 + wave * 16;

    float* bufA = lds + wave * 3 * WBUF;
    float* bufB = bufA + WBUF;
    float* bufC = bufB + WBUF;

    // Async copy mapping: one b128/lane; 8 lanes cover one 32-float row, so
    // each instruction stages 4 rows; 4 instructions stage the wave's 16 rows.
    const int ar = lane >> 3;        // row-within-quad  0..3
    const int ac = (lane & 7) * 4;   // col               0,4,..,28
    const float* gsrc = x + (size_t)(row0 + ar) * E_ + ac;

#define ISSUE_TILE(kt, buf)                                                   \
    {                                                                         \
        _Pragma("unroll")                                                     \
        for (int i = 0; i < 4; ++i) {                                         \
            const float* g = gsrc + (size_t)(4 * i) * E_ + (size_t)(kt) * KT; \
            float*       l = (buf) + (4 * i + ar) * ROWPAD + ac;              \
            GLDS_B128(g, l);                                                  \
        }                                                                     \
    }

    ISSUE_TILE(0, bufA);
    ISSUE_TILE(1, bufB);

    v8f acc0 = {};
    v8f acc1 = {};
    (void)Wm;

    for (int kt = 0; kt < NT; ++kt) {
        if (kt + 2 < NT) {
            ISSUE_TILE(kt + 2, bufC);
            __builtin_amdgcn_s_wait_asynccnt(8); // oldest 4 (tile kt) landed
        } else if (kt + 1 < NT) {
            __builtin_amdgcn_s_wait_asynccnt(4);
        } else {
            __builtin_amdgcn_s_wait_asynccnt(0);
        }

        const float* arow = bufA + n * ROWPAD + half * 2;
        const float* wk   = Wr + (size_t)(kt * (KT / 4)) * 128 + (size_t)lane * 4;
#pragma unroll
        for (int kk = 0; kk < KT; kk += 4) {
            v2f a = *(const v2f*)(arow + kk);               // ds_load_b64
            v4f wv = *(const v4f*)(wk + (kk >> 2) * 128);   // one b128: whole B data
            v2f b0 = {wv.x, wv.y};
            v2f b1 = {wv.z, wv.w};
            acc0 = __builtin_amdgcn_wmma_f32_16x16x4_f32(
                false, a, false, b0, (short)0, acc0, false, false);
            acc1 = __builtin_amdgcn_wmma_f32_16x16x4_f32(
                false, a, false, b1, (short)0, acc1, false, false);
        }

        float* t = bufA; bufA = bufB; bufB = bufC; bufC = t; // rotate ring
    }
#undef ISSUE_TILE

    const float p0 = params[n];
    const float p1 = params[n + 16];
#pragma unroll
    for (int j = 0; j < 8; ++j) {
        const int row = row0 + half * 8 + j;   // global row = b*T + t
        float s0 = __cosf(acc0[j] + p0);
        float s1 = __cosf(acc1[j] + p1);
        if (mask[row] == 0) { s0 = -1e9f; s1 = -1e9f; }
        scores[(size_t)row * H_ + n]      = s0;
        scores[(size_t)row * H_ + n + 16] = s1;
    }
}

#else // !USE_ASYNC_LDS — direct-load fallback (known-good round-1 path)

__global__ __launch_bounds__(256) void qattn_gemm_cos(
    const float* __restrict__ x, const float* __restrict__ Wm,
    const float* __restrict__ Wr, const float* __restrict__ params,
    const int* __restrict__ mask, float* __restrict__ scores)
{
    const int lane = threadIdx.x & 31;
    const int wave = threadIdx.x >> 5;
    const int half = lane >> 4;
    const int n    = lane & 15;
    const int row0 = blockIdx.x * 128 + wave * 16;

    const float* xrow = x + (size_t)(row0 + n) * E_ + half * 2;
    (void)Wm;

    v8f acc0 = {};
    v8f acc1 = {};

#pragma unroll 8
    for (int k = 0; k < E_; k += 4) {
        v2f a;
        a.x = xrow[k];
        a.y = xrow[k + 1];
        v4f wv = *(const v4f*)(Wr + (size_t)(k >> 2) * 128 + (size_t)lane * 4);
        v2f b0 = {wv.x, wv.y};
        v2f b1 = {wv.z, wv.w};
        acc0 = __builtin_amdgcn_wmma_f32_16x16x4_f32(
            false, a, false, b0, (short)0, acc0, false, false);
        acc1 = __builtin_amdgcn_wmma_f32_16x16x4_f32(
            false, a, false, b1, (short)0, acc1, false, false);
    }

    const float p0 = params[n];
    const float p1 = params[n + 16];
#pragma unroll
    for (int j = 0; j < 8; ++j) {
        const int row = row0 + half * 8 + j;
        float s0 = __cosf(acc0[j] + p0);
        float s1 = __cosf(acc1[j] + p1);
        if (mask[row] == 0) { s0 = -1e9f; s1 = -1e9f; }
        scores[(size_t)row * H_ + n]      = s0;
        scores[(size_t)row * H_ + n + 16] = s1;
    }
}

#endif // USE_ASYNC_LDS

// ---------------------------------------------------------------------------
// Kernel 2: softmax over the token axis T, independently per (b, h).
// One 256-thread block per (b, h) column of length T = 4096. Runs IN-PLACE on
// the scores buffer (each column owned by one block; elements are read before
// being overwritten by the same thread).
// ---------------------------------------------------------------------------
__global__ __launch_bounds__(256) void qattn_softmax_T(float* __restrict__ data)
{
    const int h = blockIdx.x & (H_ - 1);
    const int b = blockIdx.x >> 5;
    float* s = data + (size_t)b * T_ * H_ + h;

    __shared__ float red[8];
    const int lane = threadIdx.x & 31;
    const int wave = threadIdx.x >> 5;

    // Pass 1: max over T
    float m = -INFINITY;
    for (int t = threadIdx.x; t < T_; t += 256)
        m = fmaxf(m, s[(size_t)t * H_]);
#pragma unroll
    for (int off = 16; off > 0; off >>= 1)
        m = fmaxf(m, __shfl_xor(m, off, 32));
    if (lane == 0) red[wave] = m;
    __syncthreads();
    float bm = red[0];
#pragma unroll
    for (int i = 1; i < 8; ++i) bm = fmaxf(bm, red[i]);
    __syncthreads();

    // Pass 2: sum of exp
    float sum = 0.0f;
    for (int t = threadIdx.x; t < T_; t += 256)
        sum += __expf(s[(size_t)t * H_] - bm);
#pragma unroll
    for (int off = 16; off > 0; off >>= 1)
        sum += __shfl_xor(sum, off, 32);
    if (lane == 0) red[wave] = sum;
    __syncthreads();
    float bs = 0.0f;
#pragma unroll
    for (int i = 0; i < 8; ++i) bs += red[i];
    const float inv = 1.0f / bs;

    // Pass 3: normalize in place
    for (int t = threadIdx.x; t < T_; t += 256) {
        const size_t idx = (size_t)t * H_;
        s[idx] = __expf(s[idx] - bm) * inv;
    }
}

// ---------------------------------------------------------------------------
extern "C" void kernel_launch(void* const* d_in, const int* in_sizes, int n_in,
                              void* d_out, int out_size, void* d_ws, size_t ws_size,
                              hipStream_t stream)
{
    const float* x      = (const float*)d_in[0]; // [B,T,E] f32
    const float* Wm     = (const float*)d_in[1]; // [E,H]   f32
    const float* params = (const float*)d_in[2]; // [H,1]   f32
    const int*   mask   = (const int*)d_in[3];   // [B,T,1] i32
    float*       out    = (float*)d_out;         // [B,T,H] f32 (also scores scratch)
    float*       Wr     = (float*)d_ws;          // [E*H]   f32 swizzled W (512 KB)

    (void)in_sizes; (void)n_in; (void)out_size; (void)ws_size;

    qattn_swizzle_W<<<dim3((E_ / 4) * 32 / 256), dim3(256), 0, stream>>>(Wm, Wr);
    qattn_gemm_cos<<<dim3(M_ / 128), dim3(256), 0, stream>>>(x, Wm, Wr, params, mask, out);
    qattn_softmax_T<<<dim3(B_ * H_), dim3(256), 0, stream>>>(out);
}